// HardThresholding_83734682403206
// MI455X (gfx1250) — compile-verified
//
#include <hip/hip_runtime.h>
#include <hip/hip_bf16.h>
#include <stdint.h>

// Hard top-k thresholding per row:
//   out[r, j] = in[r, j] if |in[r,j]| is among the k largest of row r (ties: lowest index), else 0.
//
// One 1024-thread workgroup (32 wave32's) per row. The whole 128 KB row is
// async-copied into LDS once (MI455X has 320 KB LDS/WGP), then a 3-pass
// radix select (11+10+10 bits over the |x| bit pattern) runs entirely from
// LDS. HBM traffic = 1 read + 1 write of the matrix = the bandwidth floor
// (~1 GB -> ~43 us at 23.3 TB/s).

constexpr int NT  = 1024;        // threads per block
constexpr int N   = 32768;       // row length (d_sae)
constexpr int EPT = N / NT;      // 32 elements per thread
constexpr int V4  = EPT / 4;     // 8 x uint4 per thread
constexpr int CAP = 2048;        // max tracked tie elements

// Dynamic LDS layout (32-bit words):
//   [0, N)                      row bits
//   [N, N+2048)                 histogram
//   [N+2048, N+2112)            super-bin sums (64)
//   [N+2112, N+2112+CAP)        tie index list
//   [N+2112+CAP, +8)            misc: 0=kk, 1=selected bin, 2=eqCount, 3=thrIdx
constexpr int LDS_WORDS = N + 2048 + 64 + CAP + 8;

typedef unsigned uint4v __attribute__((ext_vector_type(4)));
typedef float    f4v    __attribute__((ext_vector_type(4)));

__global__ __launch_bounds__(NT, 1)
void topk_hardthresh_kernel(const float* __restrict__ in,
                            float* __restrict__ out,
                            const int* __restrict__ kptr)
{
    extern __shared__ unsigned smem[];
    unsigned* row   = smem;
    int* hist       = (int*)(smem + N);
    int* superh     = (int*)(smem + N + 2048);
    int* eqIdx      = (int*)(smem + N + 2048 + 64);
    int* misc       = (int*)(smem + N + 2048 + 64 + CAP);

    const int t = (int)threadIdx.x;
    const size_t rowBase = (size_t)blockIdx.x * (size_t)N;
    const unsigned long long gbase =
        (unsigned long long)(uintptr_t)(in + rowBase);

    // ---- Stage 0: async HBM -> LDS copy of the whole row (ASYNCcnt) ------
#pragma unroll
    for (int c = 0; c < V4; ++c) {
        const int f = c * NT + t;                               // uint4 index
        const unsigned voff    = (unsigned)(f * 16);            // byte offset
        const unsigned ldsAddr = (unsigned)(uintptr_t)(&row[4 * f]);
        asm volatile("global_load_async_to_lds_b128 %0, %1, %2"
                     :: "v"(ldsAddr), "v"(voff), "s"(gbase)
                     : "memory");
    }
    asm volatile("s_wait_asynccnt 0" ::: "memory");
    __syncthreads();

    const int k = kptr[0];
    if (t == 0) misc[0] = k;                                    // kk = k
    __syncthreads();

    const uint4v* row4 = (const uint4v*)row;

    // One radix-select pass: histogram `BINS` bins of (key >> shift) over
    // candidates satisfying (key >> prefShift) == prefVal, then locate the
    // bin containing the kk-th largest candidate. Updates misc[0] to the
    // residual rank inside the selected bin; returns the selected bin.
    auto do_pass = [&](int BINS, int shift, int prefShift,
                       unsigned prefVal) -> unsigned {
        for (int i = t; i < BINS; i += NT) hist[i] = 0;
        __syncthreads();
#pragma unroll
        for (int c = 0; c < V4; ++c) {
            const uint4v q = row4[c * NT + t];
            unsigned kx;
            kx = q.x & 0x7fffffffu;
            if ((kx >> prefShift) == prefVal) atomicAdd(&hist[(kx >> shift) & (BINS - 1)], 1);
            kx = q.y & 0x7fffffffu;
            if ((kx >> prefShift) == prefVal) atomicAdd(&hist[(kx >> shift) & (BINS - 1)], 1);
            kx = q.z & 0x7fffffffu;
            if ((kx >> prefShift) == prefVal) atomicAdd(&hist[(kx >> shift) & (BINS - 1)], 1);
            kx = q.w & 0x7fffffffu;
            if ((kx >> prefShift) == prefVal) atomicAdd(&hist[(kx >> shift) & (BINS - 1)], 1);
        }
        __syncthreads();
        const int SUP = BINS >> 5;                              // 32 bins/super
        if (t < SUP) {
            int s = 0;
#pragma unroll
            for (int i = 0; i < 32; ++i) s += hist[t * 32 + i];
            superh[t] = s;
        }
        __syncthreads();
        if (t == 0) {
            const int kk = misc[0];
            int acc = 0;
            int s = SUP - 1;
            for (; s > 0; --s) {                 // find super-bin from the top
                if (acc + superh[s] >= kk) break;
                acc += superh[s];
            }
            int b = s * 32 + 31;
            for (; b > s * 32; --b) {            // find bin inside super-bin
                if (acc + hist[b] >= kk) break;
                acc += hist[b];
            }
            misc[0] = kk - acc;                  // residual rank inside bin b
            misc[1] = b;
        }
        __syncthreads();
        return (unsigned)misc[1];
    };

    // key <= 0x7fffffff, so (key >> 31) == 0 always: pass 1 is unfiltered.
    const unsigned b1 = do_pass(2048, 20, 31, 0u);
    const unsigned b2 = do_pass(1024, 10, 20, b1);
    const unsigned p2 = (b1 << 10) | b2;
    const unsigned b3 = do_pass(1024, 0, 10, p2);
    const unsigned T  = (p2 << 10) | b3;         // exact key of k-th largest

    // ---- Tie resolution: keep first misc[0] elements with key == T -------
    if (t == 0) misc[2] = 0;
    __syncthreads();
#pragma unroll
    for (int c = 0; c < V4; ++c) {
        const int f = c * NT + t;
        const uint4v q = row4[f];
        const unsigned bb[4] = {q.x, q.y, q.z, q.w};
#pragma unroll
        for (int e = 0; e < 4; ++e) {
            if ((bb[e] & 0x7fffffffu) == T) {
                const int p = atomicAdd(&misc[2], 1);
                if (p < CAP) eqIdx[p] = 4 * f + e;
            }
        }
    }
    __syncthreads();
    if (t == 0) {
        int m = misc[2];
        if (m > CAP) m = CAP;
        for (int i = 1; i < m; ++i) {            // insertion sort (m ~ k)
            const int v = eqIdx[i];
            int j = i - 1;
            while (j >= 0 && eqIdx[j] > v) { eqIdx[j + 1] = eqIdx[j]; --j; }
            eqIdx[j + 1] = v;
        }
        int kkf = misc[0];
        if (kkf > m) kkf = m;
        if (kkf < 1) kkf = 1;
        misc[3] = eqIdx[kkf - 1];                // last kept tie index
    }
    __syncthreads();
    const int thrIdx = misc[3];

    // ---- Output pass: LDS -> HBM, non-temporal b128 stores ----------------
    f4v* orow = (f4v*)(out + rowBase);
#pragma unroll
    for (int c = 0; c < V4; ++c) {
        const int f = c * NT + t;
        const uint4v q = row4[f];
        f4v o;
        unsigned kx;
        kx = q.x & 0x7fffffffu;
        o.x = (kx > T || (kx == T && (4 * f + 0) <= thrIdx)) ? __uint_as_float(q.x) : 0.0f;
        kx = q.y & 0x7fffffffu;
        o.y = (kx > T || (kx == T && (4 * f + 1) <= thrIdx)) ? __uint_as_float(q.y) : 0.0f;
        kx = q.z & 0x7fffffffu;
        o.z = (kx > T || (kx == T && (4 * f + 2) <= thrIdx)) ? __uint_as_float(q.z) : 0.0f;
        kx = q.w & 0x7fffffffu;
        o.w = (kx > T || (kx == T && (4 * f + 3) <= thrIdx)) ? __uint_as_float(q.w) : 0.0f;
        __builtin_nontemporal_store(o, &orow[f]);
    }
}

extern "C" void kernel_launch(void* const* d_in, const int* in_sizes, int n_in,
                              void* d_out, int out_size, void* d_ws, size_t ws_size,
                              hipStream_t stream) {
    const float* in  = (const float*)d_in[0];
    const int*  kptr = (const int*)d_in[1];
    float*      out  = (float*)d_out;

    const int rows = in_sizes[0] / N;            // 4096 for the reference
    const size_t shbytes = (size_t)LDS_WORDS * sizeof(unsigned);  // ~147.7 KB

    // Allow >64 KB dynamic LDS (MI455X: 320 KB per WGP). Not a stream op,
    // safe under graph capture.
    (void)hipFuncSetAttribute((const void*)topk_hardthresh_kernel,
                              hipFuncAttributeMaxDynamicSharedMemorySize,
                              (int)shbytes);

    topk_hardthresh_kernel<<<rows, NT, shbytes, stream>>>(in, out, kptr);
}